// TGNN_63333587746843
// MI455X (gfx1250) — compile-verified
//
#include <hip/hip_runtime.h>
#include <math.h>

#define NN    20000
#define GG    100
#define EE    150000
#define EFF   400000
#define TPB   256

typedef float v2f __attribute__((ext_vector_type(2)));
typedef float v8f __attribute__((ext_vector_type(8)));

__device__ __forceinline__ float sigm(float x)  { return 1.0f / (1.0f + __expf(-x)); }
__device__ __forceinline__ float siluf(float x) { return x * sigm(x); }
__device__ __forceinline__ float mishf(float x) {
    float sp = (x > 20.0f) ? x : log1pf(__expf(x));
    return x * tanhf(sp);
}

// ---------------------------------------------------------------------------
// WMMA fp32 GEMM:  C[M x Nc] = actOut( actIn(A)[M x K] @ W[K x Nc] + b )
// block = 256 threads (8 waves). grid = (ceil(M/32), Nc/64).
// Block tile: 32 rows x 64 cols. Both operands staged in LDS (A: 32xK,
// W: Kx64), inner loop is pure ds_load + v_wmma_f32_16x16x4_f32.
// Wave w: M-subtile (w>>2)*16, N-strip (w&3)*16.
// ---------------------------------------------------------------------------
__global__ void __launch_bounds__(256)
gemm16(const float* __restrict__ A, const float* __restrict__ W,
       const float* __restrict__ bias, float* __restrict__ C,
       int M, int K, int Nc, int actIn, int actOut)
{
    __shared__ float As[32 * 256];   // 32 KB max (K <= 256)
    __shared__ float Ws[256 * 64];   // 64 KB max
    const int m0 = blockIdx.x * 32;
    const int nb = blockIdx.y * 64;

    // stage A tile (coalesced along K), fused input activation
    for (int r = 0; r < 32; ++r) {
        int row = m0 + r;
        for (int c = threadIdx.x; c < K; c += 256) {
            float av = (row < M) ? A[(size_t)row * K + c] : 0.0f;
            if (actIn == 1) av = mishf(av);
            As[r * K + c] = av;
        }
    }
    // stage W tile (coalesced): Ws[kk*64+col] = W[kk*Nc + nb + col]
    for (int idx = threadIdx.x; idx < K * 64; idx += 256) {
        int kk = idx >> 6, col = idx & 63;
        Ws[idx] = W[(size_t)kk * Nc + nb + col];
    }
    __syncthreads();

    const int lane = threadIdx.x & 31;
    const int wave = threadIdx.x >> 5;
    const int mt   = (wave >> 2) * 16;   // 0 or 16
    const int n0   = (wave & 3) * 16;    // 0,16,32,48
    const int lr   = lane & 15;          // M row (A) / N col (B,C) within tile
    const int khi  = (lane >> 4) << 1;   // lanes 16-31 hold K+2,K+3

    v8f acc = {0.f, 0.f, 0.f, 0.f, 0.f, 0.f, 0.f, 0.f};
    const float* Arow = As + (mt + lr) * K + khi;
    const float* Wc   = Ws + (khi * 64) + n0 + lr;
    for (int k = 0; k < K; k += 4) {
        v2f a; a.x = Arow[k]; a.y = Arow[k + 1];
        const float* wp = Wc + k * 64;
        v2f b; b.x = wp[0]; b.y = wp[64];
        acc = __builtin_amdgcn_wmma_f32_16x16x4_f32(false, a, false, b,
                                                    (short)0, acc, false, false);
    }

    const int rbase = (lane < 16) ? 0 : 8;
    const int col   = nb + n0 + lr;
    const float bc  = bias[col];
#pragma unroll
    for (int r = 0; r < 8; ++r) {
        int row = m0 + mt + rbase + r;
        if (row < M) {
            float v = acc[r] + bc;
            if (actOut == 1) v = siluf(v);
            C[(size_t)row * Nc + col] = v;
        }
    }
}

// ---------------------------------------------------------------------------
// Final edge readout: logits[e] = w1 . silu(W0 @ concat(nodes[fs],nodes[fd]) + b0) + b1
// block = 512 threads (16 waves), 64 edges per block (EF % 64 == 0).
// Gathered A rows (64x128), W0 (128x64) and hidden (64x64) all in LDS.
// ---------------------------------------------------------------------------
__global__ void __launch_bounds__(512)
final_edges(const float* __restrict__ nodes,
            const int* __restrict__ fs, const int* __restrict__ fd,
            const float* __restrict__ w0, const float* __restrict__ b0,
            const float* __restrict__ w1, const float* __restrict__ b1,
            float* __restrict__ logits)
{
    __shared__ float Ae[64 * 128];   // 32 KB
    __shared__ float Ws[128 * 64];   // 32 KB
    __shared__ float Hid[64 * 64];   // 16 KB
    const int e0 = blockIdx.x * 64;

    for (int idx = threadIdx.x; idx < 64 * 128; idx += 512) {
        int r = idx >> 7, c = idx & 127;
        int e = e0 + r;
        int node = (c < 64) ? fs[e] : fd[e];
        Ae[idx] = nodes[(size_t)node * 64 + (c & 63)];
    }
    for (int idx = threadIdx.x; idx < 128 * 64; idx += 512)
        Ws[idx] = w0[idx];           // [K=128][Nc=64] row-major, direct copy
    __syncthreads();

    const int lane = threadIdx.x & 31;
    const int wave = threadIdx.x >> 5;   // 0..15
    const int mt   = (wave >> 2) * 16;   // 0,16,32,48
    const int n0   = (wave & 3) * 16;    // 0,16,32,48
    const int lr   = lane & 15;
    const int khi  = (lane >> 4) << 1;

    v8f acc = {0.f, 0.f, 0.f, 0.f, 0.f, 0.f, 0.f, 0.f};
    const float* Arow = Ae + (mt + lr) * 128 + khi;
    const float* Wc   = Ws + (khi * 64) + n0 + lr;
#pragma unroll 4
    for (int k = 0; k < 128; k += 4) {
        v2f a; a.x = Arow[k]; a.y = Arow[k + 1];
        const float* wp = Wc + k * 64;
        v2f b; b.x = wp[0]; b.y = wp[64];
        acc = __builtin_amdgcn_wmma_f32_16x16x4_f32(false, a, false, b,
                                                    (short)0, acc, false, false);
    }
    const int rbase = (lane < 16) ? 0 : 8;
    const int col   = n0 + lr;
    const float bc  = b0[col];
#pragma unroll
    for (int r = 0; r < 8; ++r)
        Hid[(mt + rbase + r) * 64 + col] = siluf(acc[r] + bc);
    __syncthreads();

    if (threadIdx.x < 64) {
        float s = b1[0];
        const float* hp = Hid + threadIdx.x * 64;
#pragma unroll 8
        for (int d = 0; d < 64; ++d) s += hp[d] * w1[d];
        logits[e0 + threadIdx.x] = s;
    }
}

// --------------------------- small kernels ---------------------------------
__global__ void fillk(float* p, float v, int n) {
    int i = blockIdx.x * blockDim.x + threadIdx.x; if (i < n) p[i] = v;
}
__global__ void copyk(float* dst, const float* src, int n) {
    int i = blockIdx.x * blockDim.x + threadIdx.x; if (i < n) dst[i] = src[i];
}
__global__ void addk(float* a, const float* b, int n) {
    int i = blockIdx.x * blockDim.x + threadIdx.x; if (i < n) a[i] += b[i];
}
__global__ void concat2(const float* a, const float* b, float* out, int rows) {
    int i = blockIdx.x * blockDim.x + threadIdx.x;
    if (i < rows * 128) {
        int r = i >> 7, c = i & 127;
        out[i] = (c < 64) ? a[r * 64 + c] : b[r * 64 + (c - 64)];
    }
}
__global__ void deg_count(const int* src, float* deg, int e) {
    int i = blockIdx.x * blockDim.x + threadIdx.x;
    if (i < e) atomicAdd(&deg[src[i]], 1.0f);
}
__global__ void emb_nodes(const float* deg, const float* w, const float* b,
                          float* nodes, int n) {
    int i = blockIdx.x * blockDim.x + threadIdx.x;
    if (i < n) {
        int d = i & 63;
        float dc = fminf(deg[i >> 6], 21.0f) * (1.0f / 20.0f);
        nodes[i] = dc * w[d] + b[d];
    }
}
__global__ void sinemb(const float* t, float* se, int n) {
    int i = blockIdx.x * blockDim.x + threadIdx.x;
    if (i < n) {  // n = N*32
        int r = i >> 5, j = i & 31;
        float f = __expf(-(logf(10000.0f) / 31.0f) * (float)j);
        float a = t[r] * 31.25f * f;   // t/128*4000
        se[r * 64 + j]      = __sinf(a);
        se[r * 64 + 32 + j] = __cosf(a);
    }
}
__global__ void segsum_ab(const float* a, const float* b, const int* batch,
                          float* gsum, float* gcnt, int rows) {
    int i = blockIdx.x * blockDim.x + threadIdx.x;
    if (i < rows * 64) {
        int r = i >> 6, d = i & 63;
        int g = batch[r];
        float v = a[i] + (b ? b[i] : 0.0f);
        atomicAdd(&gsum[g * 64 + d], v);
        if (d == 0) atomicAdd(&gcnt[g], 1.0f);
    }
}
__global__ void segdiv(float* gsum, const float* gcnt, int n) {
    int i = blockIdx.x * blockDim.x + threadIdx.x;
    if (i < n) gsum[i] /= gcnt[i >> 6];
}
__global__ void gatherctx(float* ctxn, const float* ctxG, const int* batch, int n) {
    int i = blockIdx.x * blockDim.x + threadIdx.x;
    if (i < n) ctxn[i] = ctxG[batch[i >> 6] * 64 + (i & 63)];
}
__global__ void gruk(const float* gi, const float* gh, float* h, int n) {
    int i = blockIdx.x * blockDim.x + threadIdx.x;
    if (i < n) {
        int r = i >> 6, d = i & 63;
        size_t b = (size_t)r * 192;
        float rr = sigm(gi[b + d] + gh[b + d]);
        float z  = sigm(gi[b + 64 + d] + gh[b + 64 + d]);
        float nn = tanhf(gi[b + 128 + d] + rr * gh[b + 128 + d]);
        h[i] = (1.0f - z) * nn + z * h[i];
    }
}
// --------------------------- attention -------------------------------------
__global__ void attn1(const float* __restrict__ q, const float* __restrict__ k,
                      const int* src, const int* dst, float* aE, float* mN,
                      int e, int h) {
    int i = blockIdx.x * blockDim.x + threadIdx.x;
    if (i >= e * h) return;
    int ee = i / h, hh = i - ee * h;
    int s = src[ee], d = dst[ee];
    const float4* qp = (const float4*)(q + ((size_t)d * h + hh) * 64);
    const float4* kp = (const float4*)(k + ((size_t)s * h + hh) * 64);
    float acc = 0.0f;
#pragma unroll
    for (int j = 0; j < 16; ++j) {
        float4 a = qp[j], b = kp[j];
        acc += a.x * b.x + a.y * b.y + a.z * b.z + a.w * b.w;
    }
    acc *= 0.125f;                      // / sqrt(64)
    aE[i] = acc;
    atomicMax(&mN[d * h + hh], acc);    // global_atomic_max_num_f32
}
__global__ void attn2(const float* aE, const float* __restrict__ v,
                      const int* src, const int* dst, const float* mN,
                      float* den, float* outacc, int e, int h) {
    int i = blockIdx.x * blockDim.x + threadIdx.x;
    if (i >= e * h * 4) return;
    int part = i & 3, eh = i >> 2;
    int ee = eh / h, hh = eh - ee * h;
    int s = src[ee], d = dst[ee];
    float ea = __expf(aE[eh] - mN[d * h + hh]);
    if (part == 0) atomicAdd(&den[d * h + hh], ea);
    const float* vp = v + ((size_t)s * h + hh) * 64 + part * 16;
    float* op = outacc + ((size_t)d * h + hh) * 64 + part * 16;
#pragma unroll
    for (int j = 0; j < 16; ++j) atomicAdd(&op[j], ea * vp[j]);
}
__global__ void attn3(const float* outacc, const float* den, const float* sk,
                      float* nodes, int n, int h) {
    int i = blockIdx.x * blockDim.x + threadIdx.x;
    if (i < n * 64) {
        int r = i >> 6, d = i & 63;
        float s = 0.0f;
        for (int hh = 0; hh < h; ++hh)
            s += outacc[((size_t)r * h + hh) * 64 + d] / den[r * h + hh];
        s = s / (float)h + sk[i];
        nodes[i] = siluf(s);
    }
}

// ===========================================================================
extern "C" void kernel_launch(void* const* d_in, const int* in_sizes, int n_in,
                              void* d_out, int out_size, void* d_ws, size_t ws_size,
                              hipStream_t stream)
{
    const float *emb_w,*emb_b,*mlp0w,*mlp0b,*mlp1w,*mlp1b;
    const float *g0w,*g0b,*g1w,*g1b,*c0w,*c0b,*c1w,*c1b;
    const float *Lw[4][10]; // 0 time_w,1 time_b,2 q_w,3 q_b,4 k_w,5 k_b,6 v_w,7 v_b,8 sk_w,9 sk_b
    const float *wih,*whh,*bih,*bhh,*f0w,*f0b,*f1w,*f1b,*tnode;
    const int *edge,*fedge,*batch;

    auto F = [&](int i){ return (const float*)d_in[i]; };
    auto I = [&](int i){ return (const int*)d_in[i]; };

    bool sortedOrder = (n_in > 0 && in_sizes[0] == NN); // jax sorted-pytree: batch first
    if (!sortedOrder) {
        int p = 0;
        emb_w=F(p++); emb_b=F(p++);
        mlp0w=F(p++); mlp0b=F(p++); mlp1w=F(p++); mlp1b=F(p++);
        g0w=F(p++); g0b=F(p++); g1w=F(p++); g1b=F(p++);
        c0w=F(p++); c0b=F(p++); c1w=F(p++); c1b=F(p++);
        for (int l = 0; l < 4; ++l)
            for (int j = 0; j < 10; ++j) Lw[l][j] = F(p++);
        wih=F(p++); whh=F(p++); bih=F(p++); bhh=F(p++);
        f0w=F(p++); f0b=F(p++); f1w=F(p++); f1b=F(p++);
        edge=I(p++); fedge=I(p++); batch=I(p++); tnode=F(p++);
    } else {
        int p = 0;
        batch=I(p++); edge=I(p++); fedge=I(p++);
        c0b=F(p++); c0w=F(p++); c1b=F(p++); c1w=F(p++);
        emb_b=F(p++); emb_w=F(p++);
        f0b=F(p++); f0w=F(p++); f1b=F(p++); f1w=F(p++);
        g0b=F(p++); g0w=F(p++); g1b=F(p++); g1w=F(p++);
        bhh=F(p++); bih=F(p++); whh=F(p++); wih=F(p++);
        for (int l = 0; l < 4; ++l) {   // sorted keys: k,q,skip,time,v; each (b,w)
            Lw[l][5]=F(p++); Lw[l][4]=F(p++);
            Lw[l][3]=F(p++); Lw[l][2]=F(p++);
            Lw[l][9]=F(p++); Lw[l][8]=F(p++);
            Lw[l][1]=F(p++); Lw[l][0]=F(p++);
            Lw[l][7]=F(p++); Lw[l][6]=F(p++);
        }
        mlp0b=F(p++); mlp0w=F(p++); mlp1b=F(p++); mlp1w=F(p++);
        tnode=F(p++);
    }
    (void)out_size; (void)ws_size;

    // ---- workspace carve ----
    float* wsf = (float*)d_ws;
    size_t off = 0;
    auto alloc = [&](size_t n){ float* p = wsf + off; off += n; return p; };
    float* nodes   = alloc((size_t)NN*64);
    float* tbuf    = alloc((size_t)NN*64);
    float* hstate  = alloc((size_t)NN*64);
    float* ctxnode = alloc((size_t)NN*64);
    float* temb    = alloc((size_t)NN*64);
    float* sk      = alloc((size_t)NN*64);   // skip out; also sin-emb scratch
    float* nc      = alloc((size_t)NN*64);
    float* xcat    = alloc((size_t)NN*128);
    float* qb      = alloc((size_t)NN*256);
    float* kb      = alloc((size_t)NN*256);
    float* vb      = alloc((size_t)NN*256);
    float* big     = alloc((size_t)NN*256);  // mlp hidden / attention out-acc
    float* aE      = alloc((size_t)EE*4);
    float* mN      = alloc((size_t)NN*4);
    float* den     = alloc((size_t)NN*4);
    float* deg     = alloc((size_t)NN);
    float* gsum    = alloc((size_t)GG*64);
    float* gcnt    = alloc((size_t)GG);
    float* ghid    = alloc((size_t)GG*256);
    float* ctxG    = alloc((size_t)GG*64);

    auto cdiv = [](int a, int b){ return (a + b - 1) / b; };
    const int* src = edge;
    const int* dst = edge + EE;

#define GEMM(A_,W_,B_,C_,M_,K_,N_,AI_,AO_) \
    gemm16<<<dim3(cdiv(M_,32),(N_)/64), 256, 0, stream>>>(A_,W_,B_,C_,M_,K_,N_,AI_,AO_)
#define EW(kern, n, ...) kern<<<cdiv((n),TPB), TPB, 0, stream>>>(__VA_ARGS__)

    // ---- node embedding from degree ----
    EW(fillk, NN, deg, 0.0f, NN);
    EW(deg_count, EE, src, deg, EE);
    EW(emb_nodes, NN*64, deg, emb_w, emb_b, nodes, NN*64);
    EW(copyk, NN*64, hstate, nodes, NN*64);

    // ---- t = mlp2(sin_emb(t_node)) ----
    EW(sinemb, NN*32, tnode, sk, NN*32);
    GEMM(sk,  mlp0w, mlp0b, big,  NN, 64, 256, 0, 1);
    GEMM(big, mlp1w, mlp1b, tbuf, NN, 256, 64, 0, 0);

    // ---- initial contexts ----
    EW(fillk, GG*64, gsum, 0.0f, GG*64);
    EW(fillk, GG,    gcnt, 0.0f, GG);
    EW(segsum_ab, NN*64, nodes, (const float*)nullptr, batch, gsum, gcnt, NN);
    EW(segdiv, GG*64, gsum, gcnt, GG*64);
    GEMM(gsum, g0w, g0b, ghid, GG, 64, 256, 0, 1);
    GEMM(ghid, g1w, g1b, ctxG, GG, 256, 64, 0, 0);
    EW(gatherctx, NN*64, ctxnode, ctxG, batch, NN*64);

    const int heads[4] = {4, 4, 4, 1};
    for (int l = 0; l < 4; ++l) {
        const int h = heads[l];
        // t_emb = time(mish(t)); x = concat(nodes, t_emb)
        GEMM(tbuf, Lw[l][0], Lw[l][1], temb, NN, 64, 64, 1, 0);
        EW(concat2, NN*128, nodes, temb, xcat, NN);
        // q,k,v,skip projections
        GEMM(xcat, Lw[l][2], Lw[l][3], qb, NN, 128, h*64, 0, 0);
        GEMM(xcat, Lw[l][4], Lw[l][5], kb, NN, 128, h*64, 0, 0);
        GEMM(xcat, Lw[l][6], Lw[l][7], vb, NN, 128, h*64, 0, 0);
        GEMM(xcat, Lw[l][8], Lw[l][9], sk, NN, 128, 64,   0, 0);
        // segment-softmax attention
        EW(fillk, NN*h,    mN,  -3.0e38f, NN*h);
        EW(fillk, NN*h,    den, 0.0f,     NN*h);
        EW(fillk, NN*h*64, big, 0.0f,     NN*h*64);
        EW(attn1, EE*h,   qb, kb, src, dst, aE, mN, EE, h);
        EW(attn2, EE*h*4, aE, vb, src, dst, mN, den, big, EE, h);
        EW(attn3, NN*64,  big, den, sk, nodes, NN, h);
        // GRU
        GEMM(nodes,  wih, bih, qb, NN, 64, 192, 0, 0);
        GEMM(hstate, whh, bhh, kb, NN, 64, 192, 0, 0);
        EW(gruk, NN*64, qb, kb, hstate, NN*64);
        EW(copyk, NN*64, nodes, hstate, NN*64);
        // context update
        EW(concat2, NN*128, nodes, ctxnode, xcat, NN);
        GEMM(xcat, c0w, c0b, big, NN, 128, 256, 0, 1);
        GEMM(big,  c1w, c1b, nc,  NN, 256, 64,  0, 0);
        EW(fillk, GG*64, gsum, 0.0f, GG*64);
        EW(fillk, GG,    gcnt, 0.0f, GG);
        EW(segsum_ab, NN*64, ctxnode, nc, batch, gsum, gcnt, NN);
        EW(segdiv, GG*64, gsum, gcnt, GG*64);
        GEMM(gsum, g0w, g0b, ghid, GG, 64, 256, 0, 1);
        GEMM(ghid, g1w, g1b, ctxG, GG, 256, 64, 0, 0);
        EW(gatherctx, NN*64, ctxnode, ctxG, batch, NN*64);
        EW(addk, NN*64, nodes, ctxnode, NN*64);
    }

    // ---- final edge MLP (gather-GEMM) + outputs ----
    float* out = (float*)d_out;
    final_edges<<<EFF/64, 512, 0, stream>>>(nodes, fedge, fedge + EFF,
                                            f0w, f0b, f1w, f1b, out);
    EW(copyk, NN*64, out + EFF, nodes, NN*64);

#undef GEMM
#undef EW
}